// PaSP_43473658970559
// MI455X (gfx1250) — compile-verified
//
#include <hip/hip_runtime.h>

// ---------------------------------------------------------------------------
// PaSP mesh-compaction pass for gfx1250 (MI455X, wave32).
// Bandwidth-bound (~160 MB @ 23.3 TB/s => ~7us roofline). No matrix math, so
// no WMMA; CDNA5 paths used: wave32 shuffle scans, async global->LDS staging
// (ASYNCcnt), B128 loads, NT stores, global_prefetch_b8.
// ---------------------------------------------------------------------------

#define TPB   256           // 8 wave32 per block
#define ITEMS 16            // elements per thread
#define TILE  (TPB * ITEMS) // 4096 elements per block
#define NW    (TPB / 32)    // waves per block

// faces LDS tile: 48 ints per thread, +4 pad -> stride 52 words (208B, 16B
// aligned per row; gcd(52,64)=4 -> at most 2-way bank conflict on b128 reads)
#define FROW    (3 * ITEMS)   // 48
#define FPAD    4
#define FSTRIDE (FROW + FPAD) // 52

// ---- async global->LDS support (guarded; falls back to sync staging) -------
#if defined(__has_builtin)
#if __has_builtin(__builtin_amdgcn_global_load_async_to_lds_b128)
#define HAVE_ASYNC_LDS 1
#endif
#endif

#ifdef HAVE_ASYNC_LDS
// Signature learned from the compiler diagnostic: param 0 is
// '__attribute__((__vector_size__(4 * sizeof(int)))) int __device__ *'
// i.e. a v4i32* in the global (AS1) address space; LDS side is AS3.
typedef int v4i_t __attribute__((vector_size(4 * sizeof(int))));
typedef __attribute__((address_space(1))) v4i_t gbl_v4i_t;
typedef __attribute__((address_space(3))) v4i_t lds_v4i_t;
#endif

__device__ __forceinline__ void wait_async0() {
#if defined(__has_builtin)
#if __has_builtin(__builtin_amdgcn_s_wait_asynccnt)
    __builtin_amdgcn_s_wait_asynccnt(0);
#else
    asm volatile("s_wait_asynccnt 0x0" ::: "memory");
#endif
#endif
}

// ---- wave32 + block helpers ------------------------------------------------

__device__ __forceinline__ int block_reduce_sum(int c, int* sh) {
    #pragma unroll
    for (int d = 16; d > 0; d >>= 1) c += __shfl_down(c, d, 32);
    const int lane = threadIdx.x & 31, w = threadIdx.x >> 5;
    if (lane == 0) sh[w] = c;
    __syncthreads();
    int s = 0;
    if (threadIdx.x == 0) {
        #pragma unroll
        for (int i = 0; i < NW; i++) s += sh[i];
    }
    return s; // valid on thread 0 only
}

// exclusive prefix over the per-thread counts of the whole block
__device__ __forceinline__ int block_excl_scan(int c, int* sh) {
    const int lane = threadIdx.x & 31, w = threadIdx.x >> 5;
    int inc = c;
    #pragma unroll
    for (int d = 1; d < 32; d <<= 1) {
        int x = __shfl_up(inc, d, 32);
        if (lane >= d) inc += x;
    }
    if (lane == 31) sh[w] = inc;          // wave totals
    __syncthreads();
    if (w == 0) {                          // scan the NW partials in wave 0
        int v = (lane < NW) ? sh[lane] : 0;
        int s = v;
        #pragma unroll
        for (int d = 1; d < NW; d <<= 1) {
            int x = __shfl_up(s, d, 32);
            if (lane >= d) s += x;
        }
        if (lane < NW) sh[lane] = s - v;   // exclusive wave base
    }
    __syncthreads();
    return sh[w] + (inc - c);              // block-exclusive prefix of c
}

// ---- phase 1: per-tile flag sums -------------------------------------------

__global__ void pasp_vreduce(const int* __restrict__ occ, int n,
                             int* __restrict__ bsums) {
    __shared__ int sh[NW];
    const int base = blockIdx.x * TILE + threadIdx.x * ITEMS;
    int c = 0;
    if (base + ITEMS <= n) {               // single full/tail branch per thread
        #pragma unroll
        for (int k = 0; k < ITEMS; k += 4) {
            int4 v = *(const int4*)(occ + base + k);
            c += (v.x != 0) + (v.y != 0) + (v.z != 0) + (v.w != 0);
        }
    } else {
        for (int k = 0; k < ITEMS; k++)
            if (base + k < n) c += (occ[base + k] != 0);
    }
    int s = block_reduce_sum(c, sh);
    if (threadIdx.x == 0) bsums[blockIdx.x] = s;
}

__global__ void pasp_freduce(const int* __restrict__ faces, int f,
                             int* __restrict__ bsums) {
    __shared__ int sh[NW];
    const int base = blockIdx.x * TILE + threadIdx.x * ITEMS;
    int c = 0;
    if (base + ITEMS <= f) {
        #pragma unroll
        for (int k = 0; k < ITEMS; k++) c += (faces[3 * (base + k)] >= 0);
    } else {
        for (int k = 0; k < ITEMS; k++)
            if (base + k < f) c += (faces[3 * (base + k)] >= 0);
    }
    int s = block_reduce_sum(c, sh);
    if (threadIdx.x == 0) bsums[blockIdx.x] = s;
}

// ---- phase 2: scan the (<=1024) block sums, emit totals --------------------

__device__ void scan_small(const int* __restrict__ in, int* __restrict__ out,
                           int n, int* __restrict__ total, int* sh) {
    const int t = threadIdx.x;
    if (n == 0) { if (t == 0) *total = 0; __syncthreads(); return; }
    if (n <= 1024) {
        int v = (t < n) ? in[t] : 0;
        sh[t] = v;
        __syncthreads();
        for (int d = 1; d < 1024; d <<= 1) {
            int x = (t >= d) ? sh[t - d] : 0;
            __syncthreads();
            sh[t] += x;
            __syncthreads();
        }
        if (t < n) out[t] = sh[t] - v;     // exclusive
        if (t == n - 1) *total = sh[t];
    } else {                               // robust fallback (unused at N_V/N_F)
        if (t == 0) {
            int s = 0;
            for (int i = 0; i < n; i++) { out[i] = s; s += in[i]; }
            *total = s;
        }
        __syncthreads();
    }
}

__global__ void pasp_scan_sums(const int* v_sums, int* v_offs, int nvb,
                               const int* f_sums, int* f_offs, int nfb,
                               int* counts) {
    __shared__ int sh[1024];
    scan_small(v_sums, v_offs, nvb, counts + 0, sh);
    __syncthreads();
    scan_small(f_sums, f_offs, nfb, counts + 1, sh);
}

// ---- phase 3a: vertex compaction scatter -----------------------------------

__global__ void pasp_vscatter(const float* __restrict__ pts,
                              const int* __restrict__ occ,
                              const int* __restrict__ v_offs, int n,
                              float* __restrict__ outV) {
    __shared__ int sh[NW];
    const int blk  = blockIdx.x;
    const int base = blk * TILE + threadIdx.x * ITEMS;
    const bool full = (blk + 1) * TILE <= n;

    // prefetch next tile of occ (one cacheline per lane) [global_prefetch_b8]
    if (blk + 1 < (int)gridDim.x)
        __builtin_prefetch(occ + (blk + 1) * TILE + threadIdx.x * 16, 0, 0);

    int flags[ITEMS];
    int c = 0;
    if (full) {
        #pragma unroll
        for (int k = 0; k < ITEMS; k += 4) {
            int4 v = *(const int4*)(occ + base + k);
            flags[k + 0] = (v.x != 0); flags[k + 1] = (v.y != 0);
            flags[k + 2] = (v.z != 0); flags[k + 3] = (v.w != 0);
            c += flags[k] + flags[k + 1] + flags[k + 2] + flags[k + 3];
        }
    } else {
        #pragma unroll
        for (int k = 0; k < ITEMS; k++) {
            flags[k] = (base + k < n) && (occ[base + k] != 0);
            c += flags[k];
        }
    }

    int dst = v_offs[blk] + block_excl_scan(c, sh);

    // bulk-load this thread's 48 contiguous floats (12 x B128 when full)
    float pbuf[3 * ITEMS];
    const int p0 = 3 * base;
    if (full) {
        #pragma unroll
        for (int q = 0; q < (3 * ITEMS) / 4; q++) {
            float4 v = ((const float4*)(pts + p0))[q];
            pbuf[4 * q + 0] = v.x; pbuf[4 * q + 1] = v.y;
            pbuf[4 * q + 2] = v.z; pbuf[4 * q + 3] = v.w;
        }
    } else {
        for (int q = 0; q < 3 * ITEMS; q++)
            pbuf[q] = (p0 + q < 3 * n) ? pts[p0 + q] : 0.0f;
    }

    #pragma unroll
    for (int k = 0; k < ITEMS; k++) {
        if (flags[k]) {
            __builtin_nontemporal_store(pbuf[3 * k + 0], outV + 3 * dst + 0);
            __builtin_nontemporal_store(pbuf[3 * k + 1], outV + 3 * dst + 1);
            __builtin_nontemporal_store(pbuf[3 * k + 2], outV + 3 * dst + 2);
            dst++;
        }
    }
}

// ---- phase 3b: face remap + compaction scatter (async LDS staging) ---------

__global__ void pasp_fscatter(const int* __restrict__ faces,
                              const int* __restrict__ vmap,
                              const int* __restrict__ f_offs, int f, int nverts,
                              float* __restrict__ outF) {
    __shared__ int sh[NW];
    __shared__ int ftile[TPB * FSTRIDE];   // 53.2 KB padded face tile
    const int blk    = blockIdx.x;
    const int tbase3 = 3 * blk * TILE;     // first int of this tile
    const bool full  = (blk + 1) * TILE <= f;

    if (blk + 1 < (int)gridDim.x)
        __builtin_prefetch(faces + 3 * ((blk + 1) * TILE) + threadIdx.x * 16, 0, 0);

    // ---- stage the 48KB face tile into LDS (sequential global traffic) ----
#ifdef HAVE_ASYNC_LDS
    if (full) {
        #pragma unroll
        for (int q = 0; q < (3 * TILE) / (TPB * 4); q++) {     // 12 x B128/lane
            const int j = (q * TPB + threadIdx.x) * 4;         // tile int index
            const int w = (j / FROW) * FSTRIDE + (j % FROW);   // padded LDS word
            __builtin_amdgcn_global_load_async_to_lds_b128(
                (gbl_v4i_t*)(faces + tbase3 + j),
                (lds_v4i_t*)(ftile + w), 0, 0);
        }
        wait_async0();                     // own wave's ASYNCcnt -> 0
    } else
#endif
    {
        for (int j = threadIdx.x; j < 3 * TILE; j += TPB) {
            const int w = (j / FROW) * FSTRIDE + (j % FROW);
            ftile[w] = (full || tbase3 + j < 3 * f) ? faces[tbase3 + j] : -1;
        }
    }
    __syncthreads();                       // make all waves' tiles visible

    // ---- per-thread read back: 12 x ds_load_b128, 16B-aligned rows --------
    int fbuf[FROW];
    {
        const int4* row = (const int4*)(ftile + threadIdx.x * FSTRIDE);
        #pragma unroll
        for (int q = 0; q < FROW / 4; q++) {
            int4 v = row[q];
            fbuf[4 * q + 0] = v.x; fbuf[4 * q + 1] = v.y;
            fbuf[4 * q + 2] = v.z; fbuf[4 * q + 3] = v.w;
        }
    }

    int keep[ITEMS];
    int c = 0;
    #pragma unroll
    for (int k = 0; k < ITEMS; k++) {
        keep[k] = (fbuf[3 * k] >= 0);
        c += keep[k];
    }

    int dst = f_offs[blk] + block_excl_scan(c, sh);

    #pragma unroll
    for (int k = 0; k < ITEMS; k++) {
        if (keep[k]) {
            #pragma unroll
            for (int e = 0; e < 3; e++) {
                int a = fbuf[3 * k + e];
                a = min(max(a, 0), nverts - 1);        // clip like the reference
                float val = (float)vmap[a];            // gather remap
                __builtin_nontemporal_store(val, outF + 3 * dst + e);
            }
            dst++;
        }
    }
}

// ---- phase 4: padding tails + scalar counts --------------------------------

__global__ void pasp_padfill(float* __restrict__ out,
                             const int* __restrict__ counts, int n, int f) {
    const long long vlen = 3LL * n;
    const long long body = vlen + 3LL * f;
    const long long gid  = (long long)blockIdx.x * blockDim.x + threadIdx.x;
    const int nv = counts[0];
    const int nf = counts[1];
    if (gid < vlen) {
        if (gid >= 3LL * nv) __builtin_nontemporal_store(0.0f, out + gid);
    } else if (gid < body) {
        if (gid - vlen >= 3LL * nf) __builtin_nontemporal_store(-1.0f, out + gid);
    } else if (gid == body) {
        out[gid] = (float)nv;
    } else if (gid == body + 1) {
        out[gid] = (float)nf;
    }
}

// ---------------------------------------------------------------------------

extern "C" void kernel_launch(void* const* d_in, const int* in_sizes, int n_in,
                              void* d_out, int out_size, void* d_ws, size_t ws_size,
                              hipStream_t stream) {
    const float* pts   = (const float*)d_in[0];   // [N,3] f32
    const int*   faces = (const int*)d_in[1];     // [F,3] i32
    const int*   occ   = (const int*)d_in[2];     // [N]   i32
    const int*   vmap  = (const int*)d_in[3];     // [N]   i32

    const int N = in_sizes[2];
    const int F = in_sizes[1] / 3;

    float* out  = (float*)d_out;
    float* outV = out;
    float* outF = out + (size_t)3 * N;

    const int nvb = (N + TILE - 1) / TILE;        // 489 tiles
    const int nfb = (F + TILE - 1) / TILE;        // 977 tiles

    int* ws      = (int*)d_ws;                    // ~12 KB used, rewritten each call
    int* counts  = ws;                            // [0]=n_verts, [1]=n_faces
    int* v_sums  = ws + 2;
    int* v_offs  = v_sums + nvb;
    int* f_sums  = v_offs + nvb;
    int* f_offs  = f_sums + nfb;

    pasp_vreduce<<<nvb, TPB, 0, stream>>>(occ, N, v_sums);
    pasp_freduce<<<nfb, TPB, 0, stream>>>(faces, F, f_sums);
    pasp_scan_sums<<<1, 1024, 0, stream>>>(v_sums, v_offs, nvb,
                                           f_sums, f_offs, nfb, counts);
    pasp_vscatter<<<nvb, TPB, 0, stream>>>(pts, occ, v_offs, N, outV);
    pasp_fscatter<<<nfb, TPB, 0, stream>>>(faces, vmap, f_offs, F, N, outF);

    const long long tot = 3LL * N + 3LL * F + 2;
    const int pb = (int)((tot + 255) / 256);
    pasp_padfill<<<pb, 256, 0, stream>>>(out, counts, N, F);
}